// VectorQuantizer_42339787604559
// MI455X (gfx1250) — compile-verified
//
#include <hip/hip_runtime.h>

// Problem constants
#define NUM_EMB   1024
#define DIM       256
#define SPATIAL   16384      // 16*32*32
#define NROWS     131072     // 8 * SPATIAL
#define OUT_ELEMS 33554432   // 8 * 256 * SPATIAL

#define ROWS_PER_BLOCK 128
#define XS_STRIDE      260   // 256 + 4 pad: stride mod 64 == 4 -> conflict-free frag reads

typedef float v2f __attribute__((ext_vector_type(2)));
typedef float v8f __attribute__((ext_vector_type(8)));

// ---------------------------------------------------------------------------
// Kernel 1: hn[k] = 0.5 * ||emb[k]||^2 ; zero the loss accumulator
// ---------------------------------------------------------------------------
__global__ void vq_norms_kernel(const float* __restrict__ emb,
                                float* __restrict__ hn,
                                float* __restrict__ lossAcc)
{
    int k = blockIdx.x * blockDim.x + threadIdx.x;
    if (k == 0) *lossAcc = 0.0f;
    if (k < NUM_EMB) {
        const float* e = emb + (size_t)k * DIM;
        float s = 0.0f;
        for (int c = 0; c < DIM; ++c) { float v = e[c]; s += v * v; }
        hn[k] = 0.5f * s;
    }
}

// ---------------------------------------------------------------------------
// Kernel 2: fp32 WMMA distance argmin with async-DMA double-buffered codebook.
//   score(n,k) = 0.5*||e_k||^2 - x_n . e_k   (argmin == argmin of L2 distance)
// Block: 256 threads (8 waves), 128 rows. Wave w owns rows [16w, 16w+16).
// LDS: X tile [128][260] staged once (async), A panel hoisted to 128 VGPRs,
//      then the X region is dead and the two E^T buffers are overlaid on it.
// ---------------------------------------------------------------------------
__global__ __launch_bounds__(256)
void vq_argmin_kernel(const float* __restrict__ x,
                      const float* __restrict__ emb,
                      const float* __restrict__ hn,
                      int* __restrict__ idx)
{
    extern __shared__ float smem[];
    float* Xs  = smem;                      // [128][XS_STRIDE] (prologue only)
    float* Es0 = smem;                      // overlay: [16][XS_STRIDE]
    float* Es1 = smem + 16 * XS_STRIDE;     // overlay: [16][XS_STRIDE]

    const int tid    = threadIdx.x;
    const int lane   = tid & 31;
    const int wave   = tid >> 5;
    const int lm     = lane & 15;
    const bool loHalf = (lane < 16);
    const int koff   = loHalf ? 0 : 2;      // fp32 A/B: hi half-wave holds K+2,K+3

    const int rowBase = blockIdx.x * ROWS_PER_BLOCK;  // never crosses a batch
    const int b  = rowBase >> 14;
    const int s0 = rowBase & (SPATIAL - 1);
    const float* xb = x + (size_t)b * DIM * SPATIAL + s0;

    // ---- Stage X tile via async DMA: lanes sweep r -> coalesced global side
    {
        const int r  = tid & 127;
        const int c0 = (tid >> 7) * 128;
        for (int i = 0; i < 128; ++i) {
            int c = c0 + i;
            unsigned long long ga = (unsigned long long)(uintptr_t)(xb + (size_t)c * SPATIAL + r);
            unsigned int       la = (unsigned int)(uintptr_t)&Xs[r * XS_STRIDE + c];
            asm volatile("global_load_async_to_lds_b32 %0, %1, off"
                         :: "v"(la), "v"(ga) : "memory");
        }
    }
    asm volatile("s_wait_asynccnt 0x0" ::: "memory");
    __syncthreads();

    // ---- Hoist this wave's A fragments (16x256 panel) into registers -------
    const int arow = wave * 16 + lm;
    v2f afrag[64];
#pragma unroll
    for (int i = 0; i < 64; ++i)
        afrag[i] = *(const v2f*)&Xs[arow * XS_STRIDE + i * 4 + koff];

    __syncthreads();   // X tile dead everywhere -> E^T buffers may be written

    float bestv[8];
    int   besti[8];
#pragma unroll
    for (int r = 0; r < 8; ++r) { bestv[r] = 3.4e38f; besti[r] = 0; }

    const int erow = tid >> 4;          // 0..15: codeword row within tile
    const int ecol = (tid & 15) * 16;   // 16 floats (4 x b128) per thread

    // async-stage one 16x256 E^T tile (16 KB) into dst
    auto stage_async = [&](float* dst, int kt) {
        const float* src = emb + (size_t)(kt * 16 + erow) * DIM + ecol;
        float* d = dst + erow * XS_STRIDE + ecol;
#pragma unroll
        for (int j = 0; j < 4; ++j) {
            unsigned long long ga = (unsigned long long)(uintptr_t)(src + j * 4);
            unsigned int       la = (unsigned int)(uintptr_t)(d + j * 4);
            asm volatile("global_load_async_to_lds_b128 %0, %1, off"
                         :: "v"(la), "v"(ga) : "memory");
        }
    };

    stage_async(Es0, 0);   // prologue: tile 0 in flight

    for (int kt = 0; kt < 64; ++kt) {
        const float* Eb = (kt & 1) ? Es1 : Es0;
        float*       En = (kt & 1) ? Es0 : Es1;

        asm volatile("s_wait_asynccnt 0x0" ::: "memory"); // my buf[cur] loads done
        __syncthreads();   // all waves' buf[cur] written AND all done with buf[1-cur]

        if (kt + 1 < 64)
            stage_async(En, kt + 1);       // DMA next panel, overlaps the WMMAs below

        const float hnl = hn[kt * 16 + lm];

        v8f acc = {};
#pragma unroll
        for (int i = 0; i < 64; ++i) {
            v2f bfrag = *(const v2f*)&Eb[lm * XS_STRIDE + i * 4 + koff];
            acc = __builtin_amdgcn_wmma_f32_16x16x4_f32(
                false, afrag[i], false, bfrag, (short)0, acc, false, false);
        }

        const int myk = kt * 16 + lm;      // column N == lane%16
#pragma unroll
        for (int r = 0; r < 8; ++r) {      // acc[r]: row M=r (lanes 0-15) / M=8+r
            float score = hnl - acc[r];
            if (score < bestv[r]) { bestv[r] = score; besti[r] = myk; }
        }
    }

    // ---- Reduce argmin across the 16 lanes holding one row (xor<=8 stays in group)
#pragma unroll
    for (int r = 0; r < 8; ++r) {
        float v = bestv[r];
        int   bi = besti[r];
#pragma unroll
        for (int m = 8; m >= 1; m >>= 1) {
            float ov = __shfl_xor(v, m);
            int   ob = __shfl_xor(bi, m);
            if (ov < v || (ov == v && ob < bi)) { v = ov; bi = ob; } // first-min tie-break
        }
        if (lm == 0) {
            int rr = wave * 16 + r + (loHalf ? 0 : 8);
            idx[rowBase + rr] = bi;
        }
    }
}

// ---------------------------------------------------------------------------
// Kernel 3: gather codewords to output (x's layout) + fused MSE partial
// ---------------------------------------------------------------------------
__global__ __launch_bounds__(256)
void vq_gather_loss_kernel(const float* __restrict__ x,
                           const float* __restrict__ emb,
                           const int* __restrict__ idx,
                           float* __restrict__ out,
                           float* __restrict__ lossAcc)
{
    const int n = blockIdx.x * 256 + threadIdx.x;
    const int b = n >> 14;
    const int s = n & (SPATIAL - 1);
    const int code = idx[n];
    const float* e = emb + (size_t)code * DIM;
    const size_t base = (size_t)b * DIM * SPATIAL + s;

    float acc = 0.0f;
    for (int c = 0; c < DIM; ++c) {
        float q  = e[c];                      // gather: emb stays L2-resident (1 MB)
        size_t a = base + (size_t)c * SPATIAL;
        float xv = x[a];
        out[a] = q;                           // coalesced: lanes sweep s
        float d = q - xv;
        acc += d * d;
    }
#pragma unroll
    for (int m = 16; m >= 1; m >>= 1) acc += __shfl_xor(acc, m);
    if ((threadIdx.x & 31) == 0) atomicAdd(lossAcc, acc);
}

// ---------------------------------------------------------------------------
// Kernel 4: finalize loss scalar
// ---------------------------------------------------------------------------
__global__ void vq_finalize_kernel(const float* __restrict__ lossAcc,
                                   float* __restrict__ out)
{
    out[OUT_ELEMS] = 1.25f * (*lossAcc) / (float)OUT_ELEMS;
}

// ---------------------------------------------------------------------------
extern "C" void kernel_launch(void* const* d_in, const int* in_sizes, int n_in,
                              void* d_out, int out_size, void* d_ws, size_t ws_size,
                              hipStream_t stream)
{
    const float* x   = (const float*)d_in[0];   // [8,256,16,32,32] fp32
    const float* emb = (const float*)d_in[1];   // [1024,256] fp32
    float* out = (float*)d_out;                 // 33554432 quantized + 1 loss

    // workspace: hn[1024] | lossAcc | pad | idx[131072]
    float* hn      = (float*)d_ws;
    float* lossAcc = hn + NUM_EMB;
    int*   idx     = (int*)((char*)d_ws + 2048 * sizeof(float));

    vq_norms_kernel<<<(NUM_EMB + 255) / 256, 256, 0, stream>>>(emb, hn, lossAcc);

    size_t smem = (size_t)(ROWS_PER_BLOCK * XS_STRIDE) * sizeof(float); // E^T bufs overlaid
    vq_argmin_kernel<<<NROWS / ROWS_PER_BLOCK, 256, smem, stream>>>(x, emb, hn, idx);

    vq_gather_loss_kernel<<<NROWS / 256, 256, 0, stream>>>(x, emb, idx, out, lossAcc);

    vq_finalize_kernel<<<1, 1, 0, stream>>>(lossAcc, out);
}